// DPLSTMCell_10737418240032
// MI455X (gfx1250) — compile-verified
//
#include <hip/hip_runtime.h>
#include <math.h>

typedef __attribute__((ext_vector_type(2))) float v2f;
typedef __attribute__((ext_vector_type(8))) float v8f;

#define DK     1024            // K for both matvecs (D == H == 1024)
#define NGATE  4096            // 4*H gate rows
#define HOUT   1024
#define KSPLIT 8               // K-dimension split for memory-level parallelism
#define KCHUNK (DK / KSPLIT)   // 128 K-elements per wave

// partials[ks*NGATE + r] = dot(Wih[r, ks*KCHUNK : (ks+1)*KCHUNK], x[chunk])
//                        + dot(Whh[r, chunk], h[chunk])
// One wave computes 16 consecutive gate rows for one K-chunk with
// V_WMMA_F32_16X16X4_F32:
//   A (16x4): row M=0 = vector chunk v[k..k+3] (lane 0 holds K=0,1; lane 16: K=2,3)
//   B (4x16): B[kk][n] = W[r0+n][k+kk]; lane (16*g + n) holds K=2g,2g+1 for col n
//   D row M=0 (VGPR0 of lanes 0..15) accumulates the 16 dot products.
__global__ __launch_bounds__(256)
void lstm_gemv_wmma(const float* __restrict__ Wih,
                    const float* __restrict__ Whh,
                    const float* __restrict__ x,
                    const float* __restrict__ h,
                    float* __restrict__ partials)
{
    const int lane  = threadIdx.x & 31;
    const int wave  = threadIdx.x >> 5;
    const int r0    = (blockIdx.x * 8 + wave) * 16;  // first of 16 gate rows
    const int ks    = blockIdx.y;                    // K-slice index
    const int k0    = ks * KCHUNK;
    const int n     = lane & 15;                     // column (gate row) in tile
    const int khalf = (lane >> 4) * 2;               // K sub-offset: 0 or 2

    const float* wih_row = Wih + (size_t)(r0 + n) * DK + k0;
    const float* whh_row = Whh + (size_t)(r0 + n) * DK + k0;
    const float* xs = x + k0;
    const float* hs = h + k0;

    v8f acc = {};
    for (int k = 0; k < KCHUNK; k += 4) {
        // A operands: vector chunk in row M=0 only (lanes 0 and 16)
        v2f a_ih = {0.0f, 0.0f};
        v2f a_hh = {0.0f, 0.0f};
        if (n == 0) {
            a_ih.x = xs[k + khalf];
            a_ih.y = xs[k + khalf + 1];
            a_hh.x = hs[k + khalf];
            a_hh.y = hs[k + khalf + 1];
        }
        // B operands: W^T tile, two consecutive K elements per lane (b64 load)
        v2f b_ih = *(const v2f*)(wih_row + k + khalf);
        v2f b_hh = *(const v2f*)(whh_row + k + khalf);

        acc = __builtin_amdgcn_wmma_f32_16x16x4_f32(
                  false, a_ih, false, b_ih, (short)0, acc, false, false);
        acc = __builtin_amdgcn_wmma_f32_16x16x4_f32(
                  false, a_hh, false, b_hh, (short)0, acc, false, false);
    }

    // D row M=0 -> VGPR0 of lanes 0..15 (N = lane)
    if (lane < 16) {
        partials[(size_t)ks * NGATE + r0 + lane] = acc[0];
    }
}

__device__ __forceinline__ float sigmoidf_(float v) {
    return 1.0f / (1.0f + expf(-v));
}

__global__ __launch_bounds__(256)
void lstm_pointwise(const float* __restrict__ partials,
                    const float* __restrict__ bias_ih,
                    const float* __restrict__ bias_hh,
                    const float* __restrict__ c_prev,
                    float* __restrict__ out)
{
    int j = blockIdx.x * blockDim.x + threadIdx.x;
    if (j >= HOUT) return;

    float g4[4];
#pragma unroll
    for (int g = 0; g < 4; ++g) {
        int base = g * HOUT + j;
        float v = bias_ih[base] + bias_hh[base];
#pragma unroll
        for (int s = 0; s < KSPLIT; ++s) {       // fixed-order sum: deterministic
            v += partials[s * NGATE + base];
        }
        g4[g] = v;
    }

    float it = sigmoidf_(g4[0]);
    float ft = sigmoidf_(g4[1]);
    float gt = tanhf(g4[2]);
    float ot = sigmoidf_(g4[3]);

    float ct = ft * c_prev[j] + it * gt;         // c_prev row 0
    out[j]   = ot * tanhf(ct);
}

extern "C" void kernel_launch(void* const* d_in, const int* in_sizes, int n_in,
                              void* d_out, int out_size, void* d_ws, size_t ws_size,
                              hipStream_t stream) {
    const float* x_t     = (const float*)d_in[0];  // [1, B, D]   -> row 0 = first 1024
    const float* h_prev  = (const float*)d_in[1];  // [1, B, H]   -> row 0 = first 1024
    const float* c_prev  = (const float*)d_in[2];  // [B, H]      -> row 0 = first 1024
    const float* w_ih    = (const float*)d_in[3];  // [4096, 1024] row-major
    const float* w_hh    = (const float*)d_in[4];  // [4096, 1024] row-major
    const float* bias_ih = (const float*)d_in[5];  // [4096]
    const float* bias_hh = (const float*)d_in[6];  // [4096]
    float*       out     = (float*)d_out;          // [1024]

    float* partials = (float*)d_ws;                // KSPLIT * 4096 floats (128 KB)

    // 4096 rows / (16 rows/wave * 8 waves/block) = 32 row-tiles; x KSPLIT k-slices
    dim3 grid(NGATE / (16 * 8), KSPLIT);
    lstm_gemv_wmma<<<grid, 256, 0, stream>>>(w_ih, w_hh, x_t, h_prev, partials);
    lstm_pointwise<<<(HOUT + 255) / 256, 256, 0, stream>>>(partials, bias_ih, bias_hh,
                                                           c_prev, out);
}